// CrossSessionCenterAlignmentLoss_1537598292250
// MI455X (gfx1250) — compile-verified
//
#include <hip/hip_runtime.h>
#include <hip/hip_bf16.h>

typedef __attribute__((ext_vector_type(2))) float v2f;
typedef __attribute__((ext_vector_type(8))) float v8f;

#define LNUM 2
#define SNUM 4
#define NGRP 8          // L*S
#define DDIM 128
#define EPSV 1e-8f

// ws layout (floats): [0,1024) raw segment sums, [1024,2048) normalized segment
// sums, [2048,2056) counts.
#define WS_FLOATS (2 * NGRP * DDIM + NGRP)

__global__ void csca_zero_ws(float* __restrict__ ws) {
  int i = blockIdx.x * blockDim.x + threadIdx.x;
  if (i < WS_FLOATS) ws[i] = 0.0f;
}

// One-pass streaming segment reduction using V_WMMA_F32_16X16X4_F32.
// Per 4 samples: A (16x4) selector matrix, rows 0-7 = one-hot(gid) (raw sums),
// rows 8-15 = one-hot(gid) * 1/max(||f||,eps) (normalized sums).
// B (4x16) = 4 samples x 16-dim tile of features; 8 tiles cover D=128.
// C (16x16, v8f) accumulates: lanes 0-15 (rows 0-7) raw, lanes 16-31 (rows
// 8-15) normalized, N = dim-within-tile.
__global__ __launch_bounds__(256) void csca_accum(
    const float* __restrict__ feat, const int* __restrict__ labels,
    const int* __restrict__ sessions, float* __restrict__ ws, int B) {
  __shared__ float lds_acc[2 * NGRP * DDIM];
  __shared__ float lds_cnt[NGRP];
  for (int i = threadIdx.x; i < 2 * NGRP * DDIM; i += blockDim.x) lds_acc[i] = 0.0f;
  if (threadIdx.x < NGRP) lds_cnt[threadIdx.x] = 0.0f;
  __syncthreads();

  const int lane = threadIdx.x & 31;
  const int m = lane & 15;          // M row within half / N within tile
  const int hi = (lane >= 16) ? 1 : 0;
  const int wid = blockIdx.x * (blockDim.x >> 5) + (threadIdx.x >> 5);
  const int nwaves = gridDim.x * (blockDim.x >> 5);
  const int nsteps = B >> 2;        // 4 samples per step

  v8f c[8];
#pragma unroll
  for (int t = 0; t < 8; ++t)
#pragma unroll
    for (int j = 0; j < 8; ++j) c[t][j] = 0.0f;

  for (int step = wid; step < nsteps; step += nwaves) {
    // half-wave 0 handles samples {0,1}; half-wave 1 handles {2,3}
    const int sA = step * 4 + 2 * hi;
    const int sB = sA + 1;
    const float* rowA = feat + (size_t)sA * DDIM;
    const float* rowB = feat + (size_t)sB * DDIM;

    // B-matrix staging: one dim per lane per tile (lane-half = K row pair)
    float b0[8], b1[8];
#pragma unroll
    for (int t = 0; t < 8; ++t) {
      b0[t] = rowA[t * 16 + m];   // K = 0 (lanes<16) / K = 2 (lanes>=16)
      b1[t] = rowB[t * 16 + m];   // K = 1            / K = 3
    }

    // prefetch next iteration's rows (global_prefetch_b8)
    const int nstep = step + nwaves;
    if (nstep < nsteps) {
      const float* pA = feat + (size_t)(nstep * 4 + 2 * hi) * DDIM;
      __builtin_prefetch(pA + m * 8, 0, 1);
      __builtin_prefetch(pA + DDIM + m * 8, 0, 1);
    }

    // per-sample squared norms: reduce within each 16-lane half
    float pa = 0.0f, pb = 0.0f;
#pragma unroll
    for (int t = 0; t < 8; ++t) {
      pa = fmaf(b0[t], b0[t], pa);
      pb = fmaf(b1[t], b1[t], pb);
    }
#pragma unroll
    for (int sft = 1; sft < 16; sft <<= 1) {
      pa += __shfl_xor(pa, sft, 32);
      pb += __shfl_xor(pb, sft, 32);
    }
    const float rnA = 1.0f / fmaxf(sqrtf(pa), EPSV);  // torch eps clamp on ||f||
    const float rnB = 1.0f / fmaxf(sqrtf(pb), EPSV);

    const int gidA = labels[sA] * SNUM + sessions[sA];
    const int gidB = labels[sB] * SNUM + sessions[sB];

    // A-matrix values: rows 0-7 one-hot, rows 8-15 scaled one-hot
    const float a0 = (m < 8) ? ((m == gidA) ? 1.0f : 0.0f)
                             : (((m - 8) == gidA) ? rnA : 0.0f);
    const float a1 = (m < 8) ? ((m == gidB) ? 1.0f : 0.0f)
                             : (((m - 8) == gidB) ? rnB : 0.0f);
    v2f a; a.x = a0; a.y = a1;

#pragma unroll
    for (int t = 0; t < 8; ++t) {
      v2f b; b.x = b0[t]; b.y = b1[t];
      c[t] = __builtin_amdgcn_wmma_f32_16x16x4_f32(
          false, a, false, b, (short)0, c[t], false, false);
    }

    // counts: lane 0 counts samples {0,1}, lane 16 counts {2,3}
    if (m == 0) {
      unsafeAtomicAdd(&lds_cnt[gidA], 1.0f);
      unsafeAtomicAdd(&lds_cnt[gidB], 1.0f);
    }
  }

  // flush wave accumulators into block LDS (ds_add_f32)
  {
    float* dst = lds_acc + (hi ? NGRP * DDIM : 0);  // raw vs normalized half
#pragma unroll
    for (int t = 0; t < 8; ++t)
#pragma unroll
      for (int j = 0; j < 8; ++j)
        unsafeAtomicAdd(&dst[j * DDIM + t * 16 + m], c[t][j]);
  }
  __syncthreads();

  // block -> global (global_atomic_add_f32)
  for (int i = threadIdx.x; i < 2 * NGRP * DDIM; i += blockDim.x)
    unsafeAtomicAdd(&ws[i], lds_acc[i]);
  if (threadIdx.x < NGRP)
    unsafeAtomicAdd(&ws[2 * NGRP * DDIM + threadIdx.x], lds_cnt[threadIdx.x]);
}

// all threads get the block-wide sum back
__device__ __forceinline__ float block_reduce128(float v, float* red) {
  const int tid = threadIdx.x;
  red[tid] = v;
  __syncthreads();
  for (int s = 64; s > 0; s >>= 1) {
    if (tid < s) red[tid] += red[tid + s];
    __syncthreads();
  }
  float r = red[0];
  __syncthreads();
  return r;
}

__global__ __launch_bounds__(128) void csca_finalize(
    const float* __restrict__ ws, float* __restrict__ out, int B) {
  __shared__ float cmat[NGRP][DDIM];   // centers
  __shared__ float nmat[NGRP][DDIM];   // normalized-feature segment sums
  __shared__ float proto[LNUM][DDIM];
  __shared__ float red[128];
  __shared__ float cnt[NGRP];
  __shared__ float csq[NGRP];          // ||c_g||^2
  const int d = threadIdx.x;           // DDIM == blockDim.x == 128

  if (d < NGRP) cnt[d] = ws[2 * NGRP * DDIM + d];
  __syncthreads();

  for (int g = 0; g < NGRP; ++g) {
    cmat[g][d] = ws[g * DDIM + d] / cnt[g];
    nmat[g][d] = ws[NGRP * DDIM + g * DDIM + d];
  }
  for (int y = 0; y < LNUM; ++y) {
    float p = 0.0f;
    for (int s = 0; s < SNUM; ++s) p += cmat[y * SNUM + s][d];
    proto[y][d] = p * (1.0f / SNUM);
  }
  __syncthreads();

  // center loss: 1 - (1/B) * sum_g (n_g . c_g) / max(||c_g||, eps)
  float center_sum = 0.0f;
  for (int g = 0; g < NGRP; ++g) {
    const float dot = block_reduce128(nmat[g][d] * cmat[g][d], red);
    const float sq  = block_reduce128(cmat[g][d] * cmat[g][d], red);
    if (d == 0) csq[g] = sq;
    center_sum += dot / fmaxf(sqrtf(sq), EPSV);
  }
  __syncthreads();
  const float center_loss = 1.0f - center_sum / (float)B;

  // align loss with the reference's running division quirk
  float per_class[LNUM];
  for (int y = 0; y < LNUM; ++y) {
    const float sqp = block_reduce128(proto[y][d] * proto[y][d], red);
    const float np = fmaxf(sqrtf(sqp), EPSV);
    float acc = 0.0f;
    for (int s = 0; s < SNUM; ++s) {
      const int g = y * SNUM + s;
      const float dot = block_reduce128(cmat[g][d] * proto[y][d], red);
      const float nc = fmaxf(sqrtf(csq[g]), EPSV);
      acc += 1.0f - dot / (nc * np);
    }
    per_class[y] = acc;
  }
  float align = 0.0f;
  for (int y = 0; y < LNUM; ++y) align = (align + per_class[y]) * (1.0f / SNUM);

  if (d == 0) {
    out[0] = center_loss + align;  // total
    out[1] = center_loss;
    out[2] = align;
  }
}

extern "C" void kernel_launch(void* const* d_in, const int* in_sizes, int n_in,
                              void* d_out, int out_size, void* d_ws, size_t ws_size,
                              hipStream_t stream) {
  const float* feat = (const float*)d_in[0];
  const int* labels = (const int*)d_in[1];
  const int* sessions = (const int*)d_in[2];
  float* out = (float*)d_out;
  float* ws = (float*)d_ws;
  const int B = in_sizes[1];  // labels count

  csca_zero_ws<<<(WS_FLOATS + 255) / 256, 256, 0, stream>>>(ws);
  csca_accum<<<1024, 256, 0, stream>>>(feat, labels, sessions, ws, B);
  csca_finalize<<<1, 128, 0, stream>>>(ws, out, B);
}